// Attention_86655260164689
// MI455X (gfx1250) — compile-verified
//
#include <hip/hip_runtime.h>
#include <hip/hip_bf16.h>
#include <math.h>

// Problem constants (from reference)
#define BATCH 16
#define SEQ   2048
#define CIN   4
#define COUT  8
#define KW    5

typedef __attribute__((ext_vector_type(2))) float v2f;
typedef __attribute__((ext_vector_type(8))) float v8f;

// Compiler-only LDS ordering fence: per-wave LDS ops are in-order in hardware;
// this just stops the compiler from reordering across it. Zero instructions.
#define LDS_FENCE() do { asm volatile("" ::: "memory"); \
                         __builtin_amdgcn_wave_barrier(); } while (0)

// ---------------------------------------------------------------------------
// xor-shuffle max across the 16 lanes of each lane-row, on the VALU via
// v_permlane16_b32 with compile-time selectors sel[i] = i ^ m.
// ---------------------------------------------------------------------------
__device__ __forceinline__ float rowmax16(float v) {
#if __has_builtin(__builtin_amdgcn_permlane16)
  union { float f; unsigned u; } a, b;
  a.f = v;
  b.u = __builtin_amdgcn_permlane16(a.u, a.u, 0x67452301u, 0xEFCDAB89u, false, false); // xor 1
  a.f = fmaxf(a.f, b.f);
  b.u = __builtin_amdgcn_permlane16(a.u, a.u, 0x54761032u, 0xDCFE98BAu, false, false); // xor 2
  a.f = fmaxf(a.f, b.f);
  b.u = __builtin_amdgcn_permlane16(a.u, a.u, 0x32107654u, 0xBA98FEDCu, false, false); // xor 4
  a.f = fmaxf(a.f, b.f);
  b.u = __builtin_amdgcn_permlane16(a.u, a.u, 0xFEDCBA98u, 0x76543210u, false, false); // xor 8
  a.f = fmaxf(a.f, b.f);
  return a.f;
#else
  v = fmaxf(v, __shfl_xor(v, 1, 32));
  v = fmaxf(v, __shfl_xor(v, 2, 32));
  v = fmaxf(v, __shfl_xor(v, 4, 32));
  v = fmaxf(v, __shfl_xor(v, 8, 32));
  return v;
#endif
}

// ---------------------------------------------------------------------------
// Stage 1: causal convs (reflect pad) producing WMMA-friendly layouts.
//   qe : [B][S][8]   pre-scaled by (1/sqrt(8))*log2(e) so scores come out of
//                    the WMMA already in the log2 domain (exp2 softmax).
//   ke : [B][8][S]   channel-major (B-operand of Q·Kᵀ)
//   ve : [B][S][16]  channel 8 = 1.0 (row-sum rides in acc column 8), 9..15=0
// NOTE reference wiring: ve = conv(k, wv), ke = conv(v, wk).
// ---------------------------------------------------------------------------
__global__ void conv_embed_kernel(const float* __restrict__ q,
                                  const float* __restrict__ k,
                                  const float* __restrict__ v,
                                  const float* __restrict__ wq,
                                  const float* __restrict__ wk,
                                  const float* __restrict__ wv,
                                  float* __restrict__ qe,
                                  float* __restrict__ ke,
                                  float* __restrict__ ve) {
  int idx = blockIdx.x * blockDim.x + threadIdx.x;   // b*S + s
  if (idx >= BATCH * SEQ) return;
  int b = idx / SEQ;
  int s = idx - b * SEQ;

  float tq[CIN][KW], tk[CIN][KW], tv[CIN][KW];
#pragma unroll
  for (int ci = 0; ci < CIN; ++ci) {
#pragma unroll
    for (int j = 0; j < KW; ++j) {
      int p = s - (KW - 1) + j;
      p = (p < 0) ? -p : p;                  // reflect (edge excluded)
      size_t base = ((size_t)b * CIN + ci) * SEQ + p;
      tq[ci][j] = q[base];
      tk[ci][j] = k[base];
      tv[ci][j] = v[base];
    }
  }

  // 1/sqrt(COUT) softmax scale folded with log2(e) for exp2-domain softmax.
  const float QSCALE = 0.35355339059327373f * 1.4426950408889634f;

#pragma unroll
  for (int co = 0; co < COUT; ++co) {
    float aq = 0.f, ak = 0.f, av = 0.f;
#pragma unroll
    for (int ci = 0; ci < CIN; ++ci) {
#pragma unroll
      for (int j = 0; j < KW; ++j) {
        int w = (co * CIN + ci) * KW + j;
        aq = fmaf(tq[ci][j], wq[w], aq);   // qe = conv(q, wq)
        av = fmaf(tk[ci][j], wv[w], av);   // ve = conv(k, wv)  (swapped!)
        ak = fmaf(tv[ci][j], wk[w], ak);   // ke = conv(v, wk)  (swapped!)
      }
    }
    qe[(size_t)idx * 8 + co] = aq * QSCALE;
    ke[((size_t)b * COUT + co) * SEQ + s] = ak;
    ve[(size_t)idx * 16 + co] = av;
    ve[(size_t)idx * 16 + 8 + co] = (co == 0) ? 1.0f : 0.f; // ones column c=8
  }
}

// ---------------------------------------------------------------------------
// Stage 2: flash attention with V_WMMA_F32_16X16X4_F32.
// 8 independent waves/WG; each wave owns a 16-query-row tile and iterates
// 32 times over 64-key super-tiles (4 x 16x16 score tiles). The 4-step
// permlane16 row-max reduction, the alpha/m update and the acc rescale are
// amortized over all 4 tiles. Row-sum rides in acc column 8 (ones-column of
// ve). P bounces through per-wave LDS (hardware-ordered within a wave; no
// workgroup barriers). Softmax is exp2-domain (scores pre-scaled in qe).
// D/C layout: lane = N col (%16), VGPR r = row r (+8 for upper lane half).
// A layout (16x4): lane%16 = M, VGPR0 K=2h, VGPR1 K=2h+1 (h = lane/16).
// ---------------------------------------------------------------------------
#define KTILES 4                                  // 16-key tiles per iteration

__global__ void __launch_bounds__(256)
flash_attn_kernel(const float* __restrict__ qe,
                  const float* __restrict__ ke,
                  const float* __restrict__ ve,
                  const float* __restrict__ w_out,
                  const float* __restrict__ b_out,
                  float* __restrict__ out) {
  __shared__ float Pl[8][KTILES][16][17];         // per-wave, conflict-free pad

  const int wave = threadIdx.x >> 5;
  const int lane = threadIdx.x & 31;
  const int h    = lane >> 4;                     // lane half (0/1)
  const int tl   = lane & 15;                     // 0..15

  const int tilesPerBatch = SEQ / 128;            // 16
  const int b      = blockIdx.x / tilesPerBatch;
  const int s_base = (blockIdx.x % tilesPerBatch) * 128 + wave * 16;

  const float* qeb = qe + (size_t)b * SEQ * 8;
  const float* keb = ke + (size_t)b * COUT * SEQ;
  const float* veb = ve + (size_t)b * SEQ * 16;

  // A-operands of Q·Kᵀ, fixed for the whole loop.
  const int srow = s_base + tl;
  v2f aq0, aq1;
  aq0.x = qeb[(size_t)srow * 8 + 2 * h];
  aq0.y = qeb[(size_t)srow * 8 + 2 * h + 1];
  aq1.x = qeb[(size_t)srow * 8 + 4 + 2 * h];
  aq1.y = qeb[(size_t)srow * 8 + 5 + 2 * h];

  // Running row-max (log2 domain); lane-half h owns rows (r + 8h).
  float m[8];
  v8f acc = {0.f, 0.f, 0.f, 0.f, 0.f, 0.f, 0.f, 0.f};
#pragma unroll
  for (int r = 0; r < 8; ++r) m[r] = -1e30f;

  for (int t0 = 0; t0 < SEQ; t0 += 16 * KTILES) {
    // ---- issue ALL global loads for this iteration up front ----
    const float* kp = keb + t0 + tl;
    v2f bk[KTILES][2];
#pragma unroll
    for (int t = 0; t < KTILES; ++t) {
      bk[t][0].x = kp[(size_t)(2 * h) * SEQ + 16 * t];
      bk[t][0].y = kp[(size_t)(2 * h + 1) * SEQ + 16 * t];
      bk[t][1].x = kp[(size_t)(4 + 2 * h) * SEQ + 16 * t];
      bk[t][1].y = kp[(size_t)(5 + 2 * h) * SEQ + 16 * t];
    }
    v2f bv[KTILES][4];                            // ve operands for P·Vᵀ
#pragma unroll
    for (int t = 0; t < KTILES; ++t) {
#pragma unroll
      for (int kk = 0; kk < 4; ++kk) {
        const float* vp = veb + (size_t)(t0 + 16 * t + 4 * kk + 2 * h) * 16 + tl;
        bv[t][kk].x = vp[0];
        bv[t][kk].y = vp[16];
      }
    }

    // ---- KTILES 16x16 score tiles (already in log2 domain) ----
    v8f sc[KTILES];
#pragma unroll
    for (int t = 0; t < KTILES; ++t) {
      v8f z = {0.f, 0.f, 0.f, 0.f, 0.f, 0.f, 0.f, 0.f};
      z = __builtin_amdgcn_wmma_f32_16x16x4_f32(false, aq0, false, bk[t][0],
                                                (short)0, z, false, false);
      sc[t] = __builtin_amdgcn_wmma_f32_16x16x4_f32(false, aq1, false, bk[t][1],
                                                    (short)0, z, false, false);
    }

    // ---- online softmax: one permlane reduction per row per 64 keys ----
#pragma unroll
    for (int r = 0; r < 8; ++r) {
      float lm = fmaxf(fmaxf(sc[0][r], sc[1][r]), fmaxf(sc[2][r], sc[3][r]));
      float mn = fmaxf(m[r], rowmax16(lm));
      float alpha = exp2f(m[r] - mn);             // bare v_exp_f32
      m[r] = mn;
      acc[r] *= alpha;
#pragma unroll
      for (int t = 0; t < KTILES; ++t)
        Pl[wave][t][r + 8 * h][tl] = exp2f(sc[t][r] - mn);
    }
    LDS_FENCE();                                  // per-wave LDS is HW-ordered

    // ---- acc += P · Vᵀ  (KTILES x four K=4 WMMAs) ----
#pragma unroll
    for (int t = 0; t < KTILES; ++t) {
#pragma unroll
      for (int kk = 0; kk < 4; ++kk) {
        const int tt = 4 * kk + 2 * h;
        v2f ap;
        ap.x = Pl[wave][t][tl][tt];
        ap.y = Pl[wave][t][tl][tt + 1];
        acc = __builtin_amdgcn_wmma_f32_16x16x4_f32(false, ap, false, bv[t][kk],
                                                    (short)0, acc, false, false);
      }
    }
    LDS_FENCE();
  }

  // ---- epilogue: normalize by column 8, 8x8 projection + bias ----
#pragma unroll
  for (int r = 0; r < 8; ++r)
    Pl[wave][0][r + 8 * h][tl] = acc[r];          // unnormalized O | rowsum@c=8
  LDS_FENCE();

  const float linv = 1.0f / Pl[wave][0][tl][8];   // softmax denominator
  float o[8];
#pragma unroll
  for (int c = 0; c < 8; ++c) o[c] = Pl[wave][0][tl][c] * linv;

  const int sg = s_base + tl;
#pragma unroll
  for (int j = 0; j < 4; ++j) {
    const int co = 4 * h + j;                     // each lane: 4 out channels
    float vsum = b_out[co];
#pragma unroll
    for (int c = 0; c < 8; ++c)
      vsum = fmaf(o[c], w_out[co * 8 + c], vsum);
    out[((size_t)b * COUT + co) * SEQ + sg] = vsum;
  }
}

// ---------------------------------------------------------------------------
extern "C" void kernel_launch(void* const* d_in, const int* in_sizes, int n_in,
                              void* d_out, int out_size, void* d_ws, size_t ws_size,
                              hipStream_t stream) {
  (void)in_sizes; (void)n_in; (void)out_size; (void)ws_size;
  const float* q     = (const float*)d_in[0];
  const float* k     = (const float*)d_in[1];
  const float* v     = (const float*)d_in[2];
  const float* wq    = (const float*)d_in[3];
  const float* wk    = (const float*)d_in[4];
  const float* wv    = (const float*)d_in[5];
  const float* w_out = (const float*)d_in[6];
  const float* b_out = (const float*)d_in[7];
  float* out = (float*)d_out;

  // Workspace: qe [B*S*8] | ke [B*8*S] | ve [B*S*16]  => 4 MB total
  float* qe = (float*)d_ws;
  float* ke = qe + (size_t)BATCH * SEQ * 8;
  float* ve = ke + (size_t)BATCH * COUT * SEQ;

  const int convThreads = 256;
  const int convBlocks  = (BATCH * SEQ + convThreads - 1) / convThreads; // 128
  conv_embed_kernel<<<convBlocks, convThreads, 0, stream>>>(
      q, k, v, wq, wk, wv, qe, ke, ve);

  const int faBlocks = BATCH * (SEQ / 128);                             // 256
  flash_attn_kernel<<<faBlocks, 256, 0, stream>>>(
      qe, ke, ve, w_out, b_out, out);
}